// TemporalPooling_58746562675096
// MI455X (gfx1250) — compile-verified
//
#include <hip/hip_runtime.h>
#include <math.h>

// ---------------------------------------------------------------------------
// Problem constants (from reference): B=64, T=2048, D=512, K=8
// ---------------------------------------------------------------------------
constexpr int Bn = 64;
constexpr int Tn = 2048;
constexpr int Dn = 512;
constexpr int Kn = 8;

constexpr int TTILE  = 64;              // t rows per inner tile
constexpr int NSUB   = 2;               // tiles per workgroup
constexpr int TCHUNK = TTILE * NSUB;    // 128 t per workgroup
constexpr int NCHUNK = Tn / TCHUNK;     // 16 chunks per batch
constexpr int WAVES  = 4;
constexpr int BLOCK  = WAVES * 32;      // 128 threads (wave32)

typedef __attribute__((ext_vector_type(2))) float v2f;
typedef __attribute__((ext_vector_type(8))) float v8f;

// ---------------------------------------------------------------------------
// Phase 0: zero the workspace accumulators (ax[B,K,D] and asum[B,K])
// ---------------------------------------------------------------------------
__global__ void zero_ws_kernel(float* __restrict__ p, int n) {
    int i = blockIdx.x * blockDim.x + threadIdx.x;
    if (i < n) p[i] = 0.0f;
}

// ---------------------------------------------------------------------------
// Phase 1: fused logits -> softmax -> a^T x accumulation (WMMA fp32 16x16x4)
// grid = (NCHUNK, B), block = 128 (4 waves)
// ---------------------------------------------------------------------------
__global__ __launch_bounds__(BLOCK) void pool_phase1(
    const float* __restrict__ xg,      // [B, T, D]
    const float* __restrict__ attn_w,  // [K, D]
    const float* __restrict__ attn_b,  // [K]
    float*       __restrict__ ax_g,    // [B, K, D] accumulators
    float*       __restrict__ asum_g)  // [B, K]    accumulators
{
    // attn_w^T staged in LDS, padded to 16 zero rows (k>=8 -> 0) so all lanes
    // load branchlessly; row stride 516 floats keeps b64 reads conflict-free.
    __shared__ float aw_s[16][516];
    // soft-assign tile, zero-padded to 16 columns so WMMA operand lanes k>=8 read 0
    __shared__ float a_s[TTILE][16];
    __shared__ float asum_s[Kn];

    const int tid  = threadIdx.x;
    const int wave = tid >> 5;
    const int lane = tid & 31;
    const int l16  = lane & 15;       // N index (k) / M sub-index
    const int lhi  = lane >> 4;       // 0 or 1 (upper half of wave)

    const int b     = blockIdx.y;
    const int chunk = blockIdx.x;
    const int tbase = chunk * TCHUNK;

    // ---- one-time LDS init (zero-padded attn_w, zeroed a_s pad cols) ----
    for (int idx = tid; idx < 16 * Dn; idx += BLOCK) {
        const int r = idx >> 9;          // / 512
        const int c = idx & (Dn - 1);    // % 512
        aw_s[r][c] = (r < Kn) ? attn_w[r * Dn + c] : 0.0f;
    }
    for (int idx = tid; idx < TTILE * 16; idx += BLOCK) {
        (&a_s[0][0])[idx] = 0.0f;
    }
    if (tid < Kn) asum_s[tid] = 0.0f;
    __syncthreads();

    const float bias = (l16 < Kn) ? attn_b[l16] : 0.0f;

    // persistent ax^T accumulators: wave owns d-slice [wave*128, wave*128+128)
    const int dbase = wave * 128;
    v8f acc[8];
#pragma unroll
    for (int dt = 0; dt < 8; ++dt) acc[dt] = v8f{};

    for (int s = 0; s < NSUB; ++s) {
        const int t0 = tbase + s * TTILE;

        // ================= logits: [16t x 16k] += x[16t x 4d] * awT[4d x 16k]
        {
            const int trow = t0 + wave * 16 + l16;     // A-matrix row for this lane
            const float* xrow  = xg + ((size_t)b * Tn + trow) * Dn + lhi * 2;
            const float* awrow = &aw_s[l16][lhi * 2];

            v8f c;
#pragma unroll
            for (int r = 0; r < 8; ++r) c[r] = bias;

#pragma unroll 8
            for (int d = 0; d < Dn; d += 4) {
                // A 16x4 fp32: v0 = K{0,2}, v1 = K{1,3}  (branchless b64 loads)
                const v2f am = *(const v2f*)(xrow + d);
                // B 4x16 fp32: v0 = K{0,2} row (N = lane), v1 = K{1,3}
                const v2f bm = *(const v2f*)(awrow + d);
                c = __builtin_amdgcn_wmma_f32_16x16x4_f32(
                        false, am, false, bm, (short)0, c, false, false);
            }

            // C layout: vgpr r -> (M=r, N=l16) lanes 0-15, (M=r+8, N=l16) lanes 16-31
            if (l16 < Kn) {
                const int rrow = wave * 16 + lhi * 8;
#pragma unroll
                for (int r = 0; r < 8; ++r) a_s[rrow + r][l16] = c[r];
            }
        }
        __syncthreads();

        // ================= softmax over K=8 per t row (+ asum accumulation)
        if (tid < TTILE) {
            float v[Kn];
            float m = -1e30f;
#pragma unroll
            for (int k = 0; k < Kn; ++k) { v[k] = a_s[tid][k]; m = fmaxf(m, v[k]); }
            float ssum = 0.0f;
#pragma unroll
            for (int k = 0; k < Kn; ++k) { v[k] = __expf(v[k] - m); ssum += v[k]; }
            const float inv = 1.0f / ssum;
#pragma unroll
            for (int k = 0; k < Kn; ++k) {
                const float av = v[k] * inv;
                a_s[tid][k] = av;
                atomicAdd(&asum_s[k], av);     // ds_add_f32
            }
        }
        __syncthreads();

        // prefetch next tile's x rows into cache while the ax stage runs
        // out of L1/L2 (1024 lines of 128B = 128 KB tile, 8 per thread)
        if (s + 1 < NSUB) {
            const char* nxt = (const char*)(xg + ((size_t)b * Tn + t0 + TTILE) * Dn);
#pragma unroll
            for (int p = 0; p < 8; ++p) {
                __builtin_prefetch(nxt + (size_t)tid * 128 + (size_t)p * 16384, 0, 3);
            }
        }

        // ================= ax^T: [16d x 16k] += x^T[16d x 4t] * a[4t x 16k]
        for (int tc = 0; tc < TTILE; tc += 4) {
            // B = a tile: v0 = rows t{0,2}, v1 = rows t{1,3}; zero-padded cols k>=8
            v2f bm;
            bm[0] = a_s[tc + lhi * 2][l16];
            bm[1] = a_s[tc + 1 + lhi * 2][l16];

            const float* xr0 = xg + ((size_t)b * Tn + t0 + tc + lhi * 2) * Dn;
            const float* xr1 = xr0 + Dn;
#pragma unroll
            for (int dt = 0; dt < 8; ++dt) {
                const int dcol = dbase + dt * 16 + l16;
                v2f am;                        // A = x^T: coalesced 16-lane reads
                am[0] = xr0[dcol];
                am[1] = xr1[dcol];
                acc[dt] = __builtin_amdgcn_wmma_f32_16x16x4_f32(
                              false, am, false, bm, (short)0, acc[dt], false, false);
            }
        }
        __syncthreads();   // protect a_s before next tile's logits overwrite it
    }

    // ---- flush accumulators: lanes 0-15 hold M=r (d), N=l16 (k); k<8 valid ----
    if (l16 < Kn) {
        float* axb = ax_g + ((size_t)b * Kn + l16) * Dn + dbase + lhi * 8;
#pragma unroll
        for (int dt = 0; dt < 8; ++dt) {
#pragma unroll
            for (int r = 0; r < 8; ++r) {
                atomicAdd(&axb[dt * 16 + r], acc[dt][r]);   // global_atomic_add_f32
            }
        }
    }
    if (tid < Kn) atomicAdd(&asum_g[b * Kn + tid], asum_s[tid]);
}

// ---------------------------------------------------------------------------
// Phase 2: pooled = ax - asum*centers, L2-normalize over K*D per batch
// grid = B, block = 256
// ---------------------------------------------------------------------------
__global__ __launch_bounds__(256) void pool_phase2(
    const float* __restrict__ ax_g,    // [B, K, D]
    const float* __restrict__ asum_g,  // [B, K]
    const float* __restrict__ centers, // [K, D]
    float*       __restrict__ out)     // [B, K*D]
{
    __shared__ float red[8];
    __shared__ float s_asum[Kn];

    const int b   = blockIdx.x;
    const int tid = threadIdx.x;

    if (tid < Kn) s_asum[tid] = asum_g[b * Kn + tid];
    __syncthreads();

    const float* axb = ax_g + (size_t)b * Kn * Dn;
    float vals[16];
    float ss = 0.0f;
#pragma unroll
    for (int i = 0; i < 16; ++i) {
        const int idx = tid + i * 256;        // idx = k*D + d
        const int k = idx >> 9;               // / 512
        const float v = axb[idx] - s_asum[k] * centers[idx];
        vals[i] = v;
        ss += v * v;
    }
    // wave32 reduction then cross-wave via LDS
#pragma unroll
    for (int off = 16; off > 0; off >>= 1) ss += __shfl_down(ss, off, 32);
    if ((tid & 31) == 0) red[tid >> 5] = ss;
    __syncthreads();

    float total = 0.0f;
#pragma unroll
    for (int i = 0; i < 8; ++i) total += red[i];
    const float inv = 1.0f / fmaxf(sqrtf(total), 1e-12f);

    float* ob = out + (size_t)b * Kn * Dn;
#pragma unroll
    for (int i = 0; i < 16; ++i) ob[tid + i * 256] = vals[i] * inv;
}

// ---------------------------------------------------------------------------
// Launch
// ---------------------------------------------------------------------------
extern "C" void kernel_launch(void* const* d_in, const int* in_sizes, int n_in,
                              void* d_out, int out_size, void* d_ws, size_t ws_size,
                              hipStream_t stream) {
    (void)in_sizes; (void)n_in; (void)out_size; (void)ws_size;

    const float* x       = (const float*)d_in[0];  // [B,T,D]
    const float* centers = (const float*)d_in[1];  // [K,D]
    const float* attn_w  = (const float*)d_in[2];  // [K,D]
    const float* attn_b  = (const float*)d_in[3];  // [K]
    float* out = (float*)d_out;

    float* ax_g   = (float*)d_ws;                  // B*K*D floats
    float* asum_g = ax_g + (size_t)Bn * Kn * Dn;   // B*K floats

    const int nz = Bn * Kn * Dn + Bn * Kn;
    zero_ws_kernel<<<(nz + 255) / 256, 256, 0, stream>>>(ax_g, nz);

    dim3 g1(NCHUNK, Bn);
    pool_phase1<<<g1, BLOCK, 0, stream>>>(x, attn_w, attn_b, ax_g, asum_g);

    pool_phase2<<<Bn, 256, 0, stream>>>(ax_g, asum_g, centers, out);
}